// Self_Attn_39522289058199
// MI455X (gfx1250) — compile-verified
//
#include <hip/hip_runtime.h>
#include <stdint.h>

// Reference returns x verbatim (the attention result is discarded), so the
// optimal MI455X kernel is a pure HBM-rate copy (~16.8 MB => ~0.7 us at
// 23.3 TB/s). We ride the CDNA5 async global<->LDS data mover:
//   global_load_async_to_lds_b128  (ASYNCcnt++)
//   s_wait_asynccnt 0
//   global_store_async_from_lds_b128
// The 24-bit instruction offset is added to BOTH the LDS and the global
// address (CDNA5 ISA), and our LDS slot stride (THREADS*16 =
// 4096 B) equals the global stride per unroll step, so a single base VGPR
// pair + offset:4096k immediates addresses all 4 transfers per thread.

#define THREADS 256
#define UNROLL  4   // 4 x 16 B per thread; LDS per block = 16 KB

__global__ __launch_bounds__(THREADS)
void Self_Attn_copy_async_main(const float* __restrict__ src,
                               float* __restrict__ dst)
{
    __shared__ __align__(16) unsigned char smem[THREADS * UNROLL * 16];

    const unsigned tid = threadIdx.x;
    // Low 32 bits of a flat shared-aperture address = LDS byte offset.
    const unsigned lds0 = (unsigned)(uintptr_t)(&smem[0]) + tid * 16u;

    const uint64_t byteoff = (uint64_t)(blockIdx.x * (THREADS * UNROLL) + tid) * 16u;
    const uint64_t gs = (uint64_t)(uintptr_t)src + byteoff;
    const uint64_t gd = (uint64_t)(uintptr_t)dst + byteoff;

    // 4 outstanding async loads per lane, then one wave-local drain.
    asm volatile(
        "global_load_async_to_lds_b128 %0, %1, off\n\t"
        "global_load_async_to_lds_b128 %0, %1, off offset:4096\n\t"
        "global_load_async_to_lds_b128 %0, %1, off offset:8192\n\t"
        "global_load_async_to_lds_b128 %0, %1, off offset:12288\n\t"
        "s_wait_asynccnt 0x0"
        :: "v"(lds0), "v"(gs) : "memory");

    // Stream the staged tiles back out LDS -> global via the async engine.
    asm volatile(
        "global_store_async_from_lds_b128 %0, %1, off\n\t"
        "global_store_async_from_lds_b128 %0, %1, off offset:4096\n\t"
        "global_store_async_from_lds_b128 %0, %1, off offset:8192\n\t"
        "global_store_async_from_lds_b128 %0, %1, off offset:12288\n\t"
        "s_wait_asynccnt 0x0"
        :: "v"(gd), "v"(lds0) : "memory");
    // s_endpgm performs an implicit wait-idle; nothing else needed.
}

// Robustness tail (not launched for B=8,C=64,L=4096): plain scalar copy.
__global__ void Self_Attn_copy_tail(const float* __restrict__ src,
                                    float* __restrict__ dst,
                                    unsigned start, unsigned n)
{
    const unsigned i = start + blockIdx.x * blockDim.x + threadIdx.x;
    if (i < n) dst[i] = src[i];
}

extern "C" void kernel_launch(void* const* d_in, const int* in_sizes, int n_in,
                              void* d_out, int out_size, void* d_ws, size_t ws_size,
                              hipStream_t stream) {
    (void)in_sizes; (void)n_in; (void)d_ws; (void)ws_size;

    const float* x = (const float*)d_in[0];   // x: [B,C,L] float32
    float* out     = (float*)d_out;           // same shape/dtype

    const unsigned nfloats       = (unsigned)out_size;      // 2,097,152
    const unsigned nchunks       = nfloats / 4u;            // 16 B chunks
    const unsigned chunks_per_bl = THREADS * UNROLL;        // 1024
    const unsigned full_blocks   = nchunks / chunks_per_bl; // 512 (exact here)

    if (full_blocks) {
        Self_Attn_copy_async_main<<<full_blocks, THREADS, 0, stream>>>(x, out);
    }

    const unsigned done = full_blocks * chunks_per_bl * 4u;
    if (done < nfloats) {
        const unsigned rem = nfloats - done;
        Self_Attn_copy_tail<<<(rem + THREADS - 1u) / THREADS, THREADS, 0, stream>>>(
            x, out, done, nfloats);
    }
}